// GraphSAGE_111669149883
// MI455X (gfx1250) — compile-verified
//
#include <hip/hip_runtime.h>
#include <hip/hip_bf16.h>

typedef __attribute__((ext_vector_type(2))) float v2f;
typedef __attribute__((ext_vector_type(8))) float v8f;

#define NROWS   60000
#define DHID    128
#define DOUT    64
#define BN_EPS  1e-5f

__device__ __forceinline__ v8f wmma_f32(v2f a, v2f b, v8f c) {
  // V_WMMA_F32_16X16X4_F32 : D = A(16x4) * B(4x16) + C(16x16), fp32 throughout
  return __builtin_amdgcn_wmma_f32_16x16x4_f32(
      /*neg_a=*/false, a, /*neg_b=*/false, b,
      /*c_mod=*/(short)0, c, /*reuse_a=*/false, /*reuse_b=*/false);
}

// ---------------------------------------------------------------------------
// Zero fill
// ---------------------------------------------------------------------------
__global__ void zero_f32(float* __restrict__ p, int n) {
  int i = blockIdx.x * blockDim.x + threadIdx.x;
  if (i < n) p[i] = 0.0f;
}

// ---------------------------------------------------------------------------
// Edge scatter: agg[dst] += x[src]  (+ cnt[dst] += 1 on first layer)
// One wave per edge; each lane moves 4 of the 128 channels via HW fp atomics.
// ---------------------------------------------------------------------------
__global__ void scatter_kernel(const float* __restrict__ x,
                               const int* __restrict__ src,
                               const int* __restrict__ dst,
                               float* __restrict__ agg,
                               float* __restrict__ cnt,
                               int E, int addCnt) {
  int e    = blockIdx.x * 8 + (threadIdx.x >> 5);
  int lane = threadIdx.x & 31;
  if (e >= E) return;
  int s = src[e];
  int d = dst[e];
  if (addCnt && lane == 0) unsafeAtomicAdd(&cnt[d], 1.0f);
  float4 v = ((const float4*)(x + (size_t)s * DHID))[lane];
  float* ad = agg + (size_t)d * DHID + lane * 4;
  unsafeAtomicAdd(ad + 0, v.x);
  unsafeAtomicAdd(ad + 1, v.y);
  unsafeAtomicAdd(ad + 2, v.z);
  unsafeAtomicAdd(ad + 3, v.w);
}

// ---------------------------------------------------------------------------
// Fused SAGE GEMM:  y = (agg / max(cnt,1)) @ Wl + bl + xr @ Wr
// Block = 256 threads (8 waves), one 16-row M stripe, 8 N-tiles of 16.
// B fragments read from L2-resident weights inside the K loop (no spills;
// the scheduler pipelines them ~20 loads deep — proven in codegen).
// Also accumulates BN column sum / sumsq via fp atomics.
// ---------------------------------------------------------------------------
#define LDSA 260   // 16 rows x 256 K, padded: bank = (4m + k) % 64, conflict-free
__global__ void sage_gemm(const float* __restrict__ agg,
                          const float* __restrict__ cnt,
                          const float* __restrict__ xr,
                          const float* __restrict__ Wl,
                          const float* __restrict__ bl,
                          const float* __restrict__ Wr,
                          float* __restrict__ y,
                          float* __restrict__ colsum,
                          float* __restrict__ colsumsq) {
  __shared__ alignas(16) float As[16 * LDSA];

  int row0 = blockIdx.x * 16;

  // Stage A = [ agg/deg (K=0..127)  |  xr (K=128..255) ] into LDS
  for (int i = threadIdx.x; i < 16 * 64; i += 256) {
    int r  = i >> 6;     // row 0..15
    int ch = i & 63;     // float4 chunk 0..63 -> combined K col = ch*4
    int grow = row0 + r;
    float4 v;
    if (ch < 32) {
      v = ((const float4*)(agg + (size_t)grow * DHID))[ch];
      float inv = 1.0f / fmaxf(cnt[grow], 1.0f);
      v.x *= inv; v.y *= inv; v.z *= inv; v.w *= inv;
    } else {
      v = ((const float4*)(xr + (size_t)grow * DHID))[ch - 32];
    }
    *(float4*)&As[r * LDSA + ch * 4] = v;
  }
  __syncthreads();

  int wave = threadIdx.x >> 5;      // N tile 0..7
  int lane = threadIdx.x & 31;
  int m    = lane & 15;             // A row within tile
  int klo  = (lane >> 4) * 2;       // lane-group K offset
  int col  = wave * 16 + (lane & 15);

  v8f acc = {};
  // K = 0..127 against Wl
#pragma unroll
  for (int k = 0; k < 128; k += 4) {
    v2f a = *(const v2f*)&As[m * LDSA + k + klo];
    v2f b;
    b.x = Wl[(k + klo + 0) * DHID + col];
    b.y = Wl[(k + klo + 1) * DHID + col];
    acc = wmma_f32(a, b, acc);
  }
  // K = 128..255 against Wr
#pragma unroll
  for (int k = 0; k < 128; k += 4) {
    v2f a = *(const v2f*)&As[m * LDSA + 128 + k + klo];
    v2f b;
    b.x = Wr[(k + klo + 0) * DHID + col];
    b.y = Wr[(k + klo + 1) * DHID + col];
    acc = wmma_f32(a, b, acc);
  }

  // C/D layout: VGPR r -> row r + 8*(lane>=16), col = lane&15
  float bias = bl[col];
  float su = 0.0f, ss = 0.0f;
#pragma unroll
  for (int r = 0; r < 8; ++r) {
    float v = acc[r] + bias;
    int grow = row0 + r + (lane >> 4) * 8;
    y[(size_t)grow * DHID + col] = v;
    su += v;
    ss += v * v;
  }
  unsafeAtomicAdd(&colsum[col], su);
  unsafeAtomicAdd(&colsumsq[col], ss);
}

// ---------------------------------------------------------------------------
// BN (training stats over node dim) + ReLU, float4 per thread
// ---------------------------------------------------------------------------
__global__ void bn_relu(const float* __restrict__ y,
                        const float* __restrict__ colsum,
                        const float* __restrict__ colsumsq,
                        const float* __restrict__ gamma,
                        const float* __restrict__ beta,
                        float* __restrict__ h, int nvec4) {
  int i = blockIdx.x * blockDim.x + threadIdx.x;
  if (i >= nvec4) return;
  int c4 = (i & 31) * 4;                 // 128 channels / 4 = 32 vec4 per row
  float4 v = ((const float4*)y)[i];
  float t[4] = {v.x, v.y, v.z, v.w};
  const float invN = 1.0f / (float)NROWS;
#pragma unroll
  for (int j = 0; j < 4; ++j) {
    int c = c4 + j;
    float mean = colsum[c] * invN;
    float var  = colsumsq[c] * invN - mean * mean;
    float r    = rsqrtf(var + BN_EPS);
    t[j] = fmaxf((t[j] - mean) * r * gamma[c] + beta[c], 0.0f);
  }
  float4 o; o.x = t[0]; o.y = t[1]; o.z = t[2]; o.w = t[3];
  ((float4*)h)[i] = o;
}

// ---------------------------------------------------------------------------
// out = (h1 + h2) @ Wo + bo    M=60000 N=64 K=128
// Block = 256 threads (8 waves): 2 M-tiles x 4 N-tiles, 32-row stripe.
// ---------------------------------------------------------------------------
#define LDSB 132
__global__ void out_gemm(const float* __restrict__ h1,
                         const float* __restrict__ h2,
                         const float* __restrict__ Wo,
                         const float* __restrict__ bo,
                         float* __restrict__ out) {
  __shared__ alignas(16) float As[32 * LDSB];

  int row0 = blockIdx.x * 32;
  for (int i = threadIdx.x; i < 32 * 32; i += 256) {
    int r  = i >> 5;
    int ch = i & 31;
    size_t off = (size_t)(row0 + r) * DHID + ch * 4;
    float4 a = *(const float4*)(h1 + off);
    float4 b = *(const float4*)(h2 + off);
    a.x += b.x; a.y += b.y; a.z += b.z; a.w += b.w;
    *(float4*)&As[r * LDSB + ch * 4] = a;
  }
  __syncthreads();

  int wave = threadIdx.x >> 5;
  int lane = threadIdx.x & 31;
  int mt   = wave >> 2;              // 0..1
  int nt   = wave & 3;               // 0..3
  int m    = (lane & 15) + mt * 16;  // LDS row
  int klo  = (lane >> 4) * 2;
  int col  = nt * 16 + (lane & 15);

  v8f acc = {};
#pragma unroll
  for (int k = 0; k < 128; k += 4) {
    v2f a = *(const v2f*)&As[m * LDSB + k + klo];
    v2f b;
    b.x = Wo[(k + klo + 0) * DOUT + col];
    b.y = Wo[(k + klo + 1) * DOUT + col];
    acc = wmma_f32(a, b, acc);
  }

  float bias = bo[col];
#pragma unroll
  for (int r = 0; r < 8; ++r) {
    int grow = row0 + mt * 16 + r + (lane >> 4) * 8;
    out[(size_t)grow * DOUT + col] = acc[r] + bias;
  }
}

// ---------------------------------------------------------------------------
// Launch
// ---------------------------------------------------------------------------
extern "C" void kernel_launch(void* const* d_in, const int* in_sizes, int n_in,
                              void* d_out, int out_size, void* d_ws, size_t ws_size,
                              hipStream_t stream) {
  const float* x    = (const float*)d_in[0];
  const int*   ei   = (const int*)d_in[1];     // jnp int (x64 off -> int32), shape (2,E)
  const float* Wl1  = (const float*)d_in[2];
  const float* bl1  = (const float*)d_in[3];
  const float* Wr1  = (const float*)d_in[4];
  const float* g1   = (const float*)d_in[5];
  const float* be1  = (const float*)d_in[6];
  const float* Wl2  = (const float*)d_in[7];
  const float* bl2  = (const float*)d_in[8];
  const float* Wr2  = (const float*)d_in[9];
  const float* g2   = (const float*)d_in[10];
  const float* be2  = (const float*)d_in[11];
  const float* Wo   = (const float*)d_in[12];
  const float* bo   = (const float*)d_in[13];

  const int E = in_sizes[1] / 2;
  const int* src = ei;
  const int* dst = ei + E;

  // Workspace layout (floats): cnt | agg | y | h1 | h2 | stats(256)
  float* ws   = (float*)d_ws;
  const size_t FEAT = (size_t)NROWS * DHID;   // 7,680,000
  float* cnt   = ws;
  float* agg   = cnt + NROWS;
  float* y     = agg + FEAT;
  float* h1    = y   + FEAT;
  float* h2    = h1  + FEAT;
  float* stats = h2  + FEAT;          // [0..127]=sum, [128..255]=sumsq

  const int ZB = 256;
  const int featBlocks = (int)((FEAT + ZB - 1) / ZB);
  const int edgeBlocks = (E + 7) / 8;
  const int mTiles16   = NROWS / 16;  // 3750
  const int mTiles32   = NROWS / 32;  // 1875
  const int bnVec4     = (int)(FEAT / 4);
  const int bnBlocks   = (bnVec4 + ZB - 1) / ZB;

  // ---------------- Layer 1 ----------------
  zero_f32<<<(NROWS + ZB - 1) / ZB, ZB, 0, stream>>>(cnt, NROWS);
  zero_f32<<<featBlocks, ZB, 0, stream>>>(agg, (int)FEAT);
  zero_f32<<<1, ZB, 0, stream>>>(stats, 256);
  scatter_kernel<<<edgeBlocks, 256, 0, stream>>>(x, src, dst, agg, cnt, E, 1);
  sage_gemm<<<mTiles16, 256, 0, stream>>>(agg, cnt, x, Wl1, bl1, Wr1,
                                          y, stats, stats + 128);
  bn_relu<<<bnBlocks, ZB, 0, stream>>>(y, stats, stats + 128, g1, be1, h1, bnVec4);

  // ---------------- Layer 2 ----------------
  zero_f32<<<featBlocks, ZB, 0, stream>>>(agg, (int)FEAT);
  zero_f32<<<1, ZB, 0, stream>>>(stats, 256);
  scatter_kernel<<<edgeBlocks, 256, 0, stream>>>(h1, src, dst, agg, nullptr, E, 0);
  sage_gemm<<<mTiles16, 256, 0, stream>>>(agg, cnt, h1, Wl2, bl2, Wr2,
                                          y, stats, stats + 128);
  bn_relu<<<bnBlocks, ZB, 0, stream>>>(y, stats, stats + 128, g2, be2, h2, bnVec4);

  // ---------------- Output ----------------
  out_gemm<<<mTiles32, 256, 0, stream>>>(h1, h2, Wo, bo, (float*)d_out);
}